// PositionalEncoding_13245679141210
// MI455X (gfx1250) — compile-verified
//
#include <hip/hip_runtime.h>
#include <hip/hip_bf16.h>

// Problem constants (from reference): x is (32,3,64,64) -> b=32, h=w=64.
// row_embed is (64, 512). Output is (32, 512, 64, 64) float32.
#define B_DIM   32
#define F_DIM   512
#define H_DIM   64
#define W_DIM   64
#define K_DIM   64            // LEN_EMBEDDING (rows of row_embed)
#define HW      (H_DIM * W_DIM)             // 4096
#define PER_B   (F_DIM * HW)                // 2,097,152 floats = 8 MB
#define CX      (W_DIM / 2)
#define CY      (H_DIM / 2)

typedef __attribute__((ext_vector_type(2))) float v2f;
typedef __attribute__((ext_vector_type(4))) float v4f;
typedef __attribute__((ext_vector_type(8))) float v8f;

// ---------------------------------------------------------------------------
// Kernel 1: build pos[f, i, j] (the b=0 slice of out) via WMMA.
//   pos[f, i*W+j] = sum_k row_embed[k, f] * onehot(Z[i,j] == k)
// Tile: M = 16 features, N = 16 consecutive j at fixed i, K = 64 (16 x K=4).
// A (16x4 f32, 2 VGPRs): lane ln=l%16 holds row M=f; half hf=l/16 gives
//   K = 4*kk + 2*hf + {0 (v0), 1 (v1)}.
// B (4x16 f32, 2 VGPRs): lane ln holds col N=j; same K split across halves.
// D (16x16 f32, 8 VGPRs): lane ln = N, VGPR v = row M = v + 8*hf.
// One wave per tile; EXEC all-ones as WMMA requires.
// ---------------------------------------------------------------------------
__global__ __launch_bounds__(256) void pos_gather_wmma_kernel(
    const float* __restrict__ row_embed,   // [64, 512]
    float* __restrict__ out0)              // [512, 64, 64] (b=0 slice)
{
    const int lane = threadIdx.x & 31;
    const int wave = threadIdx.x >> 5;
    const int tile = blockIdx.x * 8 + wave;          // 1024 blocks * 8 waves = 8192 tiles

    const int ft = tile & 31;          // 32 feature tiles
    const int jt = (tile >> 5) & 3;    // 4 j tiles
    const int i  = tile >> 7;          // 64 rows
    const int f0 = ft * 16;
    const int j0 = jt * 16;

    const int ln = lane & 15;          // N index (j offset) / M index (feature offset)
    const int hf = lane >> 4;          // lane half selects K pair

    // Z for this lane's column j = j0 + ln (same for both halves).
    const int j = j0 + ln;
    int z = abs(CX - j) + abs(CY - i) - 1;
    z = z < 0 ? 0 : z;                 // 0..63, row index into row_embed

    v8f c = {};                        // f32 accumulator (exact: onehot is 0/1)

    #pragma unroll
    for (int kk = 0; kk < K_DIM / 4; ++kk) {
        const int kbase = kk * 4 + 2 * hf;

        // A fragment: A[m=ln][k] = row_embed[k][f0+ln]
        v2f a;
        a.x = row_embed[(size_t)kbase       * F_DIM + f0 + ln];
        a.y = row_embed[(size_t)(kbase + 1) * F_DIM + f0 + ln];

        // B fragment: B[k][n=ln] = (z == k) ? 1 : 0
        v2f b;
        b.x = (z == kbase)     ? 1.0f : 0.0f;
        b.y = (z == kbase + 1) ? 1.0f : 0.0f;

        // (neg_a, A, neg_b, B, c_mod, C, reuse_a, reuse_b)
        c = __builtin_amdgcn_wmma_f32_16x16x4_f32(
                false, a, false, b, (short)0, c, false, false);
    }

    // D[m][n] = pos[f0+m, i, j0+n]; lane = n, VGPR v -> m = v + 8*hf.
    // Each b32 store: lanes 0-15 and 16-31 each write 64 contiguous bytes.
    const size_t rowbase = (size_t)i * W_DIM + j0 + ln;
    #pragma unroll
    for (int v = 0; v < 8; ++v) {
        const int f = f0 + v + 8 * hf;
        out0[(size_t)f * HW + rowbase] = c[v];
    }
}

// ---------------------------------------------------------------------------
// Kernel 2: broadcast b=0 slice (8 MB, L2-resident) to b=1..31.
// float4 loads (cached) + non-temporal 128-bit stores (248 MB, don't pollute L2).
// Uses native clang vector type (v4f) so __builtin_nontemporal_store accepts it.
// ---------------------------------------------------------------------------
__global__ __launch_bounds__(256) void pos_broadcast_kernel(
    const float* __restrict__ src,     // b=0 slice
    float* __restrict__ dst)           // full output base
{
    const size_t idx = (size_t)blockIdx.x * blockDim.x + threadIdx.x;  // v4f index
    const v4f* __restrict__ s4 = (const v4f*)src;
    v4f* d4 = (v4f*)dst;

    const v4f v = s4[idx];
    const size_t per_b4 = PER_B / 4;   // 524,288 float4 per batch

    #pragma unroll
    for (int b = 1; b < B_DIM; ++b) {
        __builtin_nontemporal_store(v, &d4[(size_t)b * per_b4 + idx]);
    }
}

extern "C" void kernel_launch(void* const* d_in, const int* in_sizes, int n_in,
                              void* d_out, int out_size, void* d_ws, size_t ws_size,
                              hipStream_t stream) {
    (void)in_sizes; (void)n_in; (void)d_ws; (void)ws_size; (void)out_size;
    // d_in[0] = x (32,3,64,64) float32 -- only shapes matter, unused.
    // d_in[1] = row_embed (64,512) float32.
    const float* row_embed = (const float*)d_in[1];
    float* out = (float*)d_out;

    // 8192 tiles, 8 waves (tiles) per 256-thread block.
    pos_gather_wmma_kernel<<<1024, 256, 0, stream>>>(row_embed, out);

    // 524,288 float4 per batch slice -> 2048 blocks of 256 threads.
    pos_broadcast_kernel<<<2048, 256, 0, stream>>>(out, out);
}